// MonopolySumQuatEmbedding_26654567039201
// MI455X (gfx1250) — compile-verified
//
#include <hip/hip_runtime.h>
#include <stdint.h>

#define F_FIELDS   26
#define FIELD_DIM  10000
#define EMBED_DIM  32
#define BATCH      32768
#define NPAIRS     (BATCH * F_FIELDS)

typedef __attribute__((ext_vector_type(4))) unsigned u32x4;
typedef __attribute__((ext_vector_type(8))) unsigned u32x8;
typedef __attribute__((ext_vector_type(4))) float    f32x4;

__global__ __launch_bounds__(256) void monopoly_embed_kernel(
    const int*   __restrict__ x,      // (BATCH, 26) int32
    const float* __restrict__ prob,   // (26, 5) f32
    const float* __restrict__ cb0, const int* __restrict__ id0,
    const float* __restrict__ cb1, const int* __restrict__ id1,
    const float* __restrict__ cb2, const int* __restrict__ id2,
    const float* __restrict__ cb3, const int* __restrict__ id3,
    const float* __restrict__ cb4, const int* __restrict__ id4,
    float* __restrict__ out,          // (BATCH, 26, 32) f32
    int totalWaves)
{
    __shared__ float s_prob[132];     // 26*5 = 130 floats staged by TDM

    const int tid  = threadIdx.x;
    const int lane = tid & 31;
    const int warp = tid >> 5;

    // ---- Stage arch_prob (520 B) into LDS with the Tensor Data Mover ----
    if (warp == 0) {
        const uint64_t ga  = (uint64_t)(uintptr_t)prob;
        const uint32_t lds = (uint32_t)(uintptr_t)s_prob; // low 32b of flat shared = LDS byte offset
        u32x4 g0;
        g0.x = 1u;                                           // count=1 (valid), user mode
        g0.y = lds;                                          // lds_addr
        g0.z = (uint32_t)ga;                                 // global_addr[31:0]
        g0.w = ((uint32_t)(ga >> 32) & 0x01FFFFFFu)          // global_addr[56:32]
             | (2u << 30);                                   // type = 2 ("image")
        u32x8 g1;
        g1.s0 = 2u << 16;      // workgroup_mask=0, data_size=2 (4 bytes), no iterate/pad
        g1.s1 = 130u << 16;    // tensor_dim0 = 130 elements
        g1.s2 = 1u << 16;      // tensor_dim1 = 1
        g1.s3 = 130u << 16;    // tile_dim0 = 130
        g1.s4 = 0u;            // tile_dim1 = tile_dim2 = 0 (unused)
        g1.s5 = 130u;          // tensor_dim0_stride = 130
        g1.s6 = 0u;
        g1.s7 = 0u;
        asm volatile("tensor_load_to_lds %0, %1" :: "s"(g0), "s"(g1) : "memory");
        __builtin_amdgcn_s_wait_tensorcnt(0);
    }
    __syncthreads();

    // ---- 4 (b,f) pairs per wave: 8 lanes per pair, float4 per lane ----
    const int g  = lane >> 3;          // which of the wave's 4 pairs
    const int sl = lane & 7;           // sub-lane within the pair
    const int c0 = sl << 2;            // first of this lane's 4 channels
    const int m  = sl >> 1;            // quarter index: m = c0/8 (shared by all 4 channels)

    const int step  = totalWaves * 4;                       // 65536; NPAIRS = 13*step exactly
    int       base  = ((int)blockIdx.x * 8 + warp) * 4;     // wave's first pair

    for (; base < NPAIRS; base += step) {
        const int pair = base + g;
        const int f = pair >> 15;            // BATCH = 2^15; pair = f*BATCH + b
        const int b = pair & (BATCH - 1);

        // Prefetch next iteration's x elements (hides x -> idx -> cb chain).
        {
            const int np = pair + step;
            if (np < NPAIRS) {
                const int nf = np >> 15, nb = np & (BATCH - 1);
                __builtin_prefetch(&x[nb * F_FIELDS + nf], 0, 0);
            }
        }

        const int xo    = x[b * F_FIELDS + f] + f * FIELD_DIM;
        const int ibase = xo * 4 + m;        // this lane's sub-index slot
        const int pbase = f * 5;

        f32x4 acc = (f32x4)(0.0f);

#define GATHER_J(J, CB, ID, OFF)                                              \
        {                                                                     \
            const int   row = ID[ibase] + (OFF) * f;                          \
            const float p   = s_prob[pbase + (J)];                            \
            const f32x4 v   = *(const f32x4*)(CB + row * EMBED_DIM + c0);     \
            acc.x = fmaf(p, v.x, acc.x);                                      \
            acc.y = fmaf(p, v.y, acc.y);                                      \
            acc.z = fmaf(p, v.z, acc.z);                                      \
            acc.w = fmaf(p, v.w, acc.w);                                      \
        }

        GATHER_J(0, cb0, id0, 500)
        GATHER_J(1, cb1, id1, 256)
        GATHER_J(2, cb2, id2, 512)
        GATHER_J(3, cb3, id3, 1024)
        GATHER_J(4, cb4, id4, 2048)
#undef GATHER_J

        // Streaming (write-once) output: 128-bit NT store, 16B aligned.
        f32x4* op = (f32x4*)(out + (b * F_FIELDS + f) * EMBED_DIM + c0);
        __builtin_nontemporal_store(acc, op);
    }
}

extern "C" void kernel_launch(void* const* d_in, const int* in_sizes, int n_in,
                              void* d_out, int out_size, void* d_ws, size_t ws_size,
                              hipStream_t stream) {
    (void)in_sizes; (void)n_in; (void)d_ws; (void)ws_size; (void)out_size;

    const int*   x    = (const int*)  d_in[0];
    const float* prob = (const float*)d_in[1];
    // d_in[2] = flag (unused by the reference forward pass)
    const float* cb0 = (const float*)d_in[3];
    const int*   id0 = (const int*)  d_in[4];
    const float* cb1 = (const float*)d_in[5];
    const int*   id1 = (const int*)  d_in[6];
    const float* cb2 = (const float*)d_in[7];
    const int*   id2 = (const int*)  d_in[8];
    const float* cb3 = (const float*)d_in[9];
    const int*   id3 = (const int*)  d_in[10];
    const float* cb4 = (const float*)d_in[11];
    const int*   id4 = (const int*)  d_in[12];
    float* out = (float*)d_out;

    const int blocks  = 2048;
    const int threads = 256;                  // 8 wave32s per block
    const int totalWaves = blocks * (threads / 32);

    monopoly_embed_kernel<<<blocks, threads, 0, stream>>>(
        x, prob, cb0, id0, cb1, id1, cb2, id2, cb3, id3, cb4, id4, out, totalWaves);
}